// RGCN_82394652607121
// MI455X (gfx1250) — compile-verified
//
#include <hip/hip_runtime.h>
#include <cstdint>
#include <cstddef>

typedef __attribute__((ext_vector_type(2))) float v2f;
typedef __attribute__((ext_vector_type(8))) float v8f;

static constexpr int H = 64;      // hidden dim (layer-1 in & out, layer-2 in)
static constexpr int ODIM = 32;   // layer-2 out dim
static constexpr int TPB = 8;     // 16-edge sub-tiles per block (W_r amortization)

// ---------------------------------------------------------------- utilities

__global__ void zero_i32(int* __restrict__ p, int n) {
    int i = blockIdx.x * blockDim.x + threadIdx.x;
    if (i < n) p[i] = 0;
}

__global__ void zero_f32(float* __restrict__ p, size_t n) {
    size_t i = (size_t)blockIdx.x * blockDim.x + threadIdx.x;
    if (i < n) p[i] = 0.0f;
}

__global__ void histo_rel(const int* __restrict__ etype, int E, int* __restrict__ hist) {
    int i = blockIdx.x * blockDim.x + threadIdx.x;
    if (i < E) atomicAdd(&hist[etype[i]], 1);
}

// single-thread exclusive scan over relations: edge offsets, block offsets, cursors
__global__ void scan_offsets(const int* __restrict__ hist, int nRels,
                             int* __restrict__ off, int* __restrict__ tileOff,
                             int* __restrict__ cursor) {
    if (blockIdx.x == 0 && threadIdx.x == 0) {
        int e = 0, t = 0;
        for (int r = 0; r < nRels; ++r) {
            off[r] = e; tileOff[r] = t; cursor[r] = e;
            e += hist[r];
            t += (hist[r] + 16 * TPB - 1) / (16 * TPB);   // super-tiles of 16*TPB edges
        }
        off[nRels] = e; tileOff[nRels] = t;
    }
}

__global__ void build_perm(const int* __restrict__ etype, int E,
                           int* __restrict__ cursor, int* __restrict__ perm) {
    int i = blockIdx.x * blockDim.x + threadIdx.x;
    if (i < E) {
        int pos = atomicAdd(&cursor[etype[i]], 1);
        perm[pos] = i;
    }
}

// W[r] = sum_b c[r,b] * V[b]   (V flat: [b][IN*OUTD], W flat: [r][IN*OUTD])
__global__ void combine_W(const float* __restrict__ V, const float* __restrict__ c,
                          float* __restrict__ W, int nB, int INO) {
    int r = blockIdx.x;
    const float* cr = c + (size_t)r * nB;
    for (int i = threadIdx.x; i < INO; i += blockDim.x) {
        float acc = 0.0f;
        for (int b = 0; b < nB; ++b) acc += cr[b] * V[(size_t)b * INO + i];
        W[(size_t)r * INO + i] = acc;
    }
}

__global__ void bias_relu(float* __restrict__ h, const float* __restrict__ b,
                          size_t n, int mask) {
    size_t i = (size_t)blockIdx.x * blockDim.x + threadIdx.x;
    if (i < n) {
        float v = h[i] + b[i & mask];
        h[i] = v > 0.0f ? v : 0.0f;
    }
}

__global__ void init_out_bias(float* __restrict__ out, const float* __restrict__ b,
                              size_t n, int mask) {
    size_t i = (size_t)blockIdx.x * blockDim.x + threadIdx.x;
    if (i < n) out[i] = b[i & mask];
}

// ------------------------------------------------------- edge-tile WMMA GEMM
//
// One block owns a relation-aligned "super-tile" of up to 16*TPB edges, all
// with the same etype r. W_r (64 x OUTD) is staged in LDS ONCE (transposed,
// pitch 68 -> conflict-free 8B ds_loads), then TPB sub-tiles of 16 edges are
// processed: gather norm-scaled x[src] rows -> A(16x64), compute
// D(16 x OUTD) via OUTD/16 waves x 16 chained v_wmma_f32_16x16x4_f32 (K=64),
// then atomically scatter rows into OUT[dst]. Sub-tile loop and early-exit
// are block-uniform, so EXEC is all-ones at every WMMA.
template <int OUTD>
__global__ __launch_bounds__(32 * (OUTD / 16))
void rgcn_edge_tile_gemm(const float* __restrict__ X,      // N x 64
                         const float* __restrict__ W,      // nRels x 64 x OUTD
                         const int* __restrict__ srcIdx,
                         const int* __restrict__ dstIdx,
                         const float* __restrict__ norm,
                         const int* __restrict__ perm,
                         const int* __restrict__ off,      // nRels+1 edge offsets
                         const int* __restrict__ tileOff,  // nRels+1 super-tile offsets
                         float* __restrict__ OUT,          // N x OUTD (accumulated)
                         int nRels) {
    constexpr int NT = OUTD / 16;   // waves per block == n-tiles
    constexpr int WP = 68;          // LDS pitch in floats (conflict-free, 8B aligned)
    __shared__ float sWt[OUTD * WP];   // W_r transposed: sWt[o*WP + k] = W_r[k][o]
    __shared__ float sA[16 * WP];      // 16 edge rows x 64 features (norm-scaled)
    __shared__ int   sDst[16];
    __shared__ int   sSrc[16];
    __shared__ float sNrm[16];

    const int t = blockIdx.x;
    if (t >= tileOff[nRels]) return;           // block-uniform: EXEC stays full

    // binary search: largest r with tileOff[r] <= t
    int lo = 0, hi = nRels;
    while (hi - lo > 1) {
        int mid = (lo + hi) >> 1;
        if (tileOff[mid] <= t) lo = mid; else hi = mid;
    }
    const int r = lo;
    const int segBase = off[r] + (t - tileOff[r]) * 16 * TPB;
    const int eEnd = off[r + 1];

    const int tid = threadIdx.x;

    // stage W_r transposed into LDS (once per block, reused for TPB sub-tiles)
    const float* __restrict__ Wr = W + (size_t)r * 64 * OUTD;
    for (int i = tid; i < 64 * OUTD; i += 32 * NT) {
        int k = i / OUTD, o = i % OUTD;
        sWt[o * WP + k] = Wr[i];
    }

    const int lane = tid & 31;
    const int wave = tid >> 5;       // n-tile index
    const int n16 = lane & 15;
    const int khalf = lane >> 4;     // 0: K pair {0,1}; 1: K pair {2,3}
    const float* aBase = sA + n16 * WP;                   // A row M = lane&15
    const float* bBase = sWt + (wave * 16 + n16) * WP;    // B column N = wave*16 + n16
    const int n = wave * 16 + n16;
    const int mBase = khalf * 8;

    for (int st = 0; st < TPB; ++st) {
        const int eBase = segBase + st * 16;
        if (eBase >= eEnd) break;               // block-uniform exit

        __syncthreads();   // sWt ready (st==0) / previous sub-tile fully consumed
        // edge metadata (padded rows: norm=0, dst=-1 -> contribute nothing)
        if (tid < 16) {
            int e = eBase + tid;
            int s = 0, d = -1;
            float nm = 0.0f;
            if (e < eEnd) {
                int ei = perm[e];
                s = srcIdx[ei]; d = dstIdx[ei]; nm = norm[ei];
            }
            sSrc[tid] = s; sDst[tid] = d; sNrm[tid] = nm;
        }
        __syncthreads();
        // gather + norm-prescale the 16 source feature rows
        for (int i = tid; i < 16 * 64; i += 32 * NT) {
            int row = i >> 6, k = i & 63;
            sA[row * WP + k] = X[(size_t)sSrc[row] * 64 + k] * sNrm[row];
        }
        __syncthreads();

        v8f acc = {0.f, 0.f, 0.f, 0.f, 0.f, 0.f, 0.f, 0.f};
#pragma unroll
        for (int k0 = 0; k0 < 64; k0 += 4) {
            v2f a = *(const v2f*)(aBase + k0 + khalf * 2);   // A[M][k0+kh*2 .. +1]
            v2f b = *(const v2f*)(bBase + k0 + khalf * 2);   // B[k0+kh*2 .. +1][N]
            acc = __builtin_amdgcn_wmma_f32_16x16x4_f32(
                /*neg_a=*/false, a, /*neg_b=*/false, b,
                /*c_mod=*/(short)0, acc, /*reuse_a=*/false, /*reuse_b=*/false);
        }

        // C/D layout: VGPR j -> M = j (lanes 0-15) / M = 8+j (lanes 16-31); N = lane&15
#pragma unroll
        for (int j = 0; j < 8; ++j) {
            int d = sDst[mBase + j];
            if (d >= 0) atomicAdd(&OUT[(size_t)d * OUTD + n], acc[j]);
        }
    }
}

// ------------------------------------------------------------------ launcher

extern "C" void kernel_launch(void* const* d_in, const int* in_sizes, int n_in,
                              void* d_out, int out_size, void* d_ws, size_t ws_size,
                              hipStream_t stream) {
    const int*   src   = (const int*)d_in[0];
    const int*   dst   = (const int*)d_in[1];
    const int*   etype = (const int*)d_in[2];
    const float* norm  = (const float*)d_in[3];
    const float* emb   = (const float*)d_in[4];
    const float* V1    = (const float*)d_in[5];
    const float* c1    = (const float*)d_in[6];
    const float* b1    = (const float*)d_in[7];
    const float* V2    = (const float*)d_in[8];
    const float* c2    = (const float*)d_in[9];
    const float* b2    = (const float*)d_in[10];

    const int E     = in_sizes[0];
    const int N     = in_sizes[4] / H;
    const int nB    = in_sizes[5] / (H * H);
    const int nRels = in_sizes[6] / nB;
    float* out = (float*)d_out;

    // carve workspace (256 B aligned slabs)
    uintptr_t p = (uintptr_t)d_ws;
    auto carve = [&](size_t bytes) -> void* {
        uintptr_t q = p;
        p += (bytes + 255) & ~(size_t)255;
        return (void*)q;
    };
    int*   hist    = (int*)carve((size_t)nRels * 4);
    int*   off     = (int*)carve((size_t)(nRels + 1) * 4);
    int*   tileOff = (int*)carve((size_t)(nRels + 1) * 4);
    int*   cursor  = (int*)carve((size_t)nRels * 4);
    int*   perm    = (int*)carve((size_t)E * 4);
    float* W1      = (float*)carve((size_t)nRels * H * H * 4);
    float* W2      = (float*)carve((size_t)nRels * H * ODIM * 4);
    float* h1      = (float*)carve((size_t)N * H * 4);

    // 1) counting sort of edges by relation
    zero_i32<<<(nRels + 255) / 256, 256, 0, stream>>>(hist, nRels);
    histo_rel<<<(E + 255) / 256, 256, 0, stream>>>(etype, E, hist);
    scan_offsets<<<1, 1, 0, stream>>>(hist, nRels, off, tileOff, cursor);
    build_perm<<<(E + 255) / 256, 256, 0, stream>>>(etype, E, cursor, perm);

    // 2) fold basis decomposition into dense per-relation matrices
    combine_W<<<nRels, 256, 0, stream>>>(V1, c1, W1, nB, H * H);
    combine_W<<<nRels, 256, 0, stream>>>(V2, c2, W2, nB, H * ODIM);

    const int maxTiles = (E + 16 * TPB - 1) / (16 * TPB) + nRels;
    const size_t n1 = (size_t)N * H;
    const size_t n2 = (size_t)N * ODIM;

    // 3) layer 1: h1 = segment_sum((emb[src] @ W1_r) * norm) ; then relu(h1 + b1)
    zero_f32<<<(unsigned)((n1 + 255) / 256), 256, 0, stream>>>(h1, n1);
    rgcn_edge_tile_gemm<H><<<maxTiles, 32 * (H / 16), 0, stream>>>(
        emb, W1, src, dst, norm, perm, off, tileOff, h1, nRels);
    bias_relu<<<(unsigned)((n1 + 255) / 256), 256, 0, stream>>>(h1, b1, n1, H - 1);

    // 4) layer 2: out = b2 + segment_sum((h1[src] @ W2_r) * norm)
    init_out_bias<<<(unsigned)((n2 + 255) / 256), 256, 0, stream>>>(out, b2, n2, ODIM - 1);
    rgcn_edge_tile_gemm<ODIM><<<maxTiles, 32 * (ODIM / 16), 0, stream>>>(
        h1, W2, src, dst, norm, perm, off, tileOff, out, nRels);
}